// DynamicFilterLayer2D_82592221102678
// MI455X (gfx1250) — compile-verified
//
#include <hip/hip_runtime.h>

// Dynamic filter layer (per-pixel 3x3 filters), fp32.
// Memory-bound: ~704 MB traffic -> stream filters/output non-temporally,
// stage x tiles (with halo) into LDS via CDNA5 async global->LDS loads.

constexpr int B = 8, C = 32, H = 256, W = 256;
constexpr int HW = H * W;
constexpr int BDIM = 256;          // 8 wave32
constexpr int TILE_H = 16;
constexpr int TILE_W = 64;         // 16 thread-groups x 4 px
constexpr int LDS_W = TILE_W + 2;  // 66 valid columns (halo)
constexpr int LDS_STRIDE = 68;     // padded: 272 B rows, 16B-aligned
constexpr int LDS_H = TILE_H + 2;  // 18 rows (halo)
constexpr int LDS_ELEMS = LDS_H * LDS_W;  // 1188 floats to stage

typedef float v4f __attribute__((ext_vector_type(4)));
typedef float v2f __attribute__((ext_vector_type(2)));

typedef __attribute__((address_space(1))) int  g_i32;   // global int
typedef __attribute__((address_space(3))) int  l_i32;   // LDS int

__device__ __forceinline__ void async_ld_f32(void* lds_slot, const float* gsrc) {
#if __has_builtin(__builtin_amdgcn_global_load_async_to_lds_b32)
  __builtin_amdgcn_global_load_async_to_lds_b32((g_i32*)gsrc, (l_i32*)lds_slot, 0, 0);
#else
  asm volatile("global_load_async_to_lds_b32 %0, %1, off"
               :
               : "v"((l_i32*)lds_slot), "v"((g_i32*)gsrc)
               : "memory");
#endif
}

__device__ __forceinline__ void wait_async0() {
#if __has_builtin(__builtin_amdgcn_s_wait_asynccnt)
  __builtin_amdgcn_s_wait_asynccnt(0);
#else
  asm volatile("s_wait_asynccnt 0x0" ::: "memory");
#endif
}

__global__ __launch_bounds__(BDIM) void dynfilter3x3_kernel(
    const float* __restrict__ x, const float* __restrict__ f,
    float* __restrict__ out) {
  __shared__ __align__(16) float tile[LDS_H * LDS_STRIDE];

  const int bx  = blockIdx.x;   // 0..3   : W tiles
  const int by  = blockIdx.y;   // 0..15  : H tiles
  const int bc  = blockIdx.z;   // 0..255 : b*C + c
  const int tid = threadIdx.x;

  const int w0base = bx * TILE_W;
  const int h0     = by * TILE_H;

  const float* xplane = x + (size_t)bc * HW;

  // ---- stage x tile (+halo) into LDS via async global->LDS; zero-fill OOB
  for (int s = tid; s < LDS_ELEMS; s += BDIM) {
    const int r  = s / LDS_W;
    const int c  = s - r * LDS_W;
    const int gh = h0 - 1 + r;
    const int gw = w0base - 1 + c;
    float* slot = &tile[r * LDS_STRIDE + c];
    if ((unsigned)gh < (unsigned)H && (unsigned)gw < (unsigned)W) {
      async_ld_f32(slot, xplane + gh * W + gw);
    } else {
      *slot = 0.0f;  // zero padding of the reference
    }
  }
  wait_async0();
  __syncthreads();

  // ---- per-thread: 4 adjacent outputs at (h, w0..w0+3)
  const int tx = tid & 15;
  const int ty = tid >> 4;
  const int h  = h0 + ty;
  const int w0 = w0base + tx * 4;

  // 3 rows x 6 cols of x from LDS (b128 + b64 reads, conflict-free)
  float xr[3][6];
#pragma unroll
  for (int i = 0; i < 3; ++i) {
    const float* rp = &tile[(ty + i) * LDS_STRIDE + tx * 4];
    v4f a  = *(const v4f*)rp;
    v2f b2 = *(const v2f*)(rp + 4);
    xr[i][0] = a.x; xr[i][1] = a.y; xr[i][2] = a.z; xr[i][3] = a.w;
    xr[i][4] = b2.x; xr[i][5] = b2.y;
  }

  // filters: 9 planes, streamed once -> non-temporal b128 loads
  const float* fbase = f + (size_t)bc * 9 * HW + (size_t)h * W + w0;
  v4f acc = {0.0f, 0.0f, 0.0f, 0.0f};
#pragma unroll
  for (int k = 0; k < 9; ++k) {
    const v4f fk = __builtin_nontemporal_load((const v4f*)(fbase + (size_t)k * HW));
    const int i = k / 3;
    const int j = k - 3 * i;
    acc.x += fk.x * xr[i][j + 0];
    acc.y += fk.y * xr[i][j + 1];
    acc.z += fk.z * xr[i][j + 2];
    acc.w += fk.w * xr[i][j + 3];
  }

  v4f* op = (v4f*)(out + (size_t)bc * HW + (size_t)h * W + w0);
  __builtin_nontemporal_store(acc, op);
}

extern "C" void kernel_launch(void* const* d_in, const int* in_sizes, int n_in,
                              void* d_out, int out_size, void* d_ws, size_t ws_size,
                              hipStream_t stream) {
  const float* x = (const float*)d_in[0];   // (B, C, H, W) fp32
  const float* f = (const float*)d_in[1];   // (B, C*9, H, W) fp32
  float* out = (float*)d_out;               // (B, C, H, W) fp32
  (void)in_sizes; (void)n_in; (void)out_size; (void)d_ws; (void)ws_size;

  dim3 grid(W / TILE_W, H / TILE_H, B * C);
  dynfilter3x3_kernel<<<grid, BDIM, 0, stream>>>(x, f, out);
}